// ModiPixelContrastLossCos_80762565034251
// MI455X (gfx1250) — compile-verified
//
#include <hip/hip_runtime.h>
#include <hip/hip_bf16.h>
#include <math.h>

// ---------------------------------------------------------------------------
// ModiPixelContrastLossCos on MI455X (gfx1250, wave32)
//
// Roofline: 384 rows x 131072 f32 = 201 MB streamed once; at 23.3 TB/s the
// floor is ~8.6 us. Needed FLOPs ~0.3 GFLOP -> purely bandwidth bound.
// Each row needs {row.a1, row.a2, ||row||^2}; we compute them as entries of a
// 16x16 Gram matrix per tile (rows = [anchor1, anchor2, 14 data rows]) using
// V_WMMA_F32_16X16X4_F32 accumulating over K = D. For a Gram product the f32
// A (16x4) and B (4x16) per-lane layouts coincide (lane m: row/col m K-pair,
// lane m+16: next K-pair), so SRC0 == SRC1 with no data shuffling.
// ---------------------------------------------------------------------------

typedef __attribute__((ext_vector_type(2))) float v2f;
typedef __attribute__((ext_vector_type(8))) float v8f;

#define DDIM   131072
#define NROWS  384            // 64 pos1 + 64 pos2 + 256 neg
#define DPT    14             // data rows per 16-row tile (rows 0,1 = anchors)
#define NTILES 28             // ceil(384/14)
#define KSPLITS 32
#define THREADS 256
#define WAVES   (THREADS / 32)
#define CHUNK   (DDIM / (KSPLITS * WAVES))   // 512 floats = 128 wmma steps

#define TEMP_INV (1.0f / 0.07f)
#define EPS 1e-8f

// ws layout: NROWS rows * 4 floats: [dot_a1, dot_a2, norm2, pad]

__global__ void __launch_bounds__(THREADS)
zero_ws(float* __restrict__ ws, int n) {
  for (int i = threadIdx.x; i < n; i += THREADS) ws[i] = 0.0f;
}

__global__ void __launch_bounds__(THREADS)
gram_stage1(const float* __restrict__ pos1, const float* __restrict__ pos2,
            const float* __restrict__ neg, float* __restrict__ ws) {
  const int tid   = threadIdx.x;
  const int wave  = tid >> 5;
  const int lane  = tid & 31;
  const int m     = lane & 15;   // matrix row (A) == matrix col (B) this lane owns
  const int hi    = lane >> 4;   // 0: K-pair (0,1); 1: K-pair (2,3)
  const int tile  = blockIdx.x / KSPLITS;
  const int split = blockIdx.x % KSPLITS;

  // Per-lane row base pointer. Tile rows: 0 -> anchor1 (pos1[63]),
  // 1 -> anchor2 (pos2[63]), 2..15 -> global data rows tile*14 .. tile*14+13.
  const float* base;
  if (m == 0) {
    base = pos1 + (size_t)63 * DDIM;
  } else if (m == 1) {
    base = pos2 + (size_t)63 * DDIM;
  } else {
    int grow = tile * DPT + (m - 2);
    int r = (grow < NROWS) ? grow : 0;   // pad rows read valid memory; dropped
    if (r < 64)       base = pos1 + (size_t)r * DDIM;
    else if (r < 128) base = pos2 + (size_t)(r - 64) * DDIM;
    else              base = neg  + (size_t)(r - 128) * DDIM;
  }

  const int k0 = (split * WAVES + wave) * CHUNK;
  const float* p = base + k0 + 2 * hi;

  // G(16x16) += tile(16x4) * tile^T(4x16), K advancing by 4 per wmma.
  v8f acc = {};
#pragma unroll 8
  for (int i = 0; i < CHUNK / 4; ++i) {
    v2f a = *reinterpret_cast<const v2f*>(p);      // global_load_b64, 16B/row/step
    acc = __builtin_amdgcn_wmma_f32_16x16x4_f32(
        /*neg_a=*/false, a, /*neg_b=*/false, a,
        /*c_mod=*/(short)0, acc, /*reuse_a=*/false, /*reuse_b=*/false);
    p += 4;
  }

  float ar[8];
#pragma unroll
  for (int v = 0; v < 8; ++v) ar[v] = acc[v];

  // C/D layout: VGPR v of lane L holds G[M][N], M = v + 8*hi, N = L&15.
  // Column N=0 -> dot with anchor1 ; column N=1 -> dot with anchor2.
  // Skip destinations M<2 (anchor rows): anchors get their ws entries from
  // their natural data-row appearance (tiles 4 and 9), avoiding double count.
  if (m < 2) {
#pragma unroll
    for (int v = 0; v < 8; ++v) {
      int M = v + 8 * hi;
      if (M >= 2) {
        int g = tile * DPT + (M - 2);
        if (g < NROWS) atomicAdd(&ws[g * 4 + m], ar[v]);
      }
    }
  }
  // Diagonal G[M][M]: lanes 0..7 hold M=lane in VGPR lane; lanes 24..31 hold
  // M=lane-16 in VGPR lane-24.
  int Md = -1; float dval = 0.0f;
  if (hi == 0 && m < 8)       { Md = m; dval = ar[m]; }
  else if (hi == 1 && m >= 8) { Md = m; dval = ar[m - 8]; }
  if (Md >= 2) {
    int g = tile * DPT + (Md - 2);
    if (g < NROWS) atomicAdd(&ws[g * 4 + 2], dval);
  }
}

__device__ __forceinline__ float block_sum(float v) {
  __shared__ float red[WAVES];
#pragma unroll
  for (int o = 16; o > 0; o >>= 1) v += __shfl_xor(v, o, 32);
  const int w = threadIdx.x >> 5, l = threadIdx.x & 31;
  __syncthreads();
  if (l == 0) red[w] = v;
  __syncthreads();
  float t = (l < WAVES) ? red[l] : 0.0f;
#pragma unroll
  for (int o = WAVES / 2; o > 0; o >>= 1) t += __shfl_xor(t, o, 32);
  return t;  // lanes 0..7 of each wave read the partials -> total only in those lanes, others get 0
}

__global__ void __launch_bounds__(THREADS)
finalize(const float* __restrict__ ws, float* __restrict__ out) {
  const float n1 = fmaxf(sqrtf(ws[63 * 4 + 2]), EPS);   // ||anchor1||
  const float n2 = fmaxf(sqrtf(ws[127 * 4 + 2]), EPS);  // ||anchor2||

  // Pass 1: S1 = sum over {pos2 all, neg} of exp(cos(a1,row)/T)
  //         S2 = sum over {pos1 all, neg} of exp(cos(a2,row)/T)
  float s1 = 0.0f, s2 = 0.0f;
  for (int r = threadIdx.x; r < NROWS; r += THREADS) {
    float nr = fmaxf(sqrtf(ws[r * 4 + 2]), EPS);
    float c1 = ws[r * 4 + 0] / (n1 * nr);
    float c2 = ws[r * 4 + 1] / (n2 * nr);
    if (r >= 64)             s1 += expf(c1 * TEMP_INV);
    if (r < 64 || r >= 128)  s2 += expf(c2 * TEMP_INV);
  }
  const float S1 = block_sum(s1);
  const float S2 = block_sum(s2);

  // Pass 2: loss = sum_i -log(e1_i/(e1_i+S1)) + sum_i -log(e2_i/(e2_i+S2))
  float l = 0.0f;
  for (int r = threadIdx.x; r < NROWS; r += THREADS) {
    float nr = fmaxf(sqrtf(ws[r * 4 + 2]), EPS);
    if (r < 63) {                                   // pos1[0..62] vs anchor1
      float c1 = ws[r * 4 + 0] / (n1 * nr);
      float e1 = expf(c1 * TEMP_INV);
      l += -logf(e1 / (e1 + S1));
    }
    if (r >= 64 && r < 127) {                       // pos2[0..62] vs anchor2
      float c2 = ws[r * 4 + 1] / (n2 * nr);
      float e2 = expf(c2 * TEMP_INV);
      l += -logf(e2 / (e2 + S2));
    }
  }
  const float L = block_sum(l);
  if (threadIdx.x == 0) out[0] = L;
}

extern "C" void kernel_launch(void* const* d_in, const int* in_sizes, int n_in,
                              void* d_out, int out_size, void* d_ws, size_t ws_size,
                              hipStream_t stream) {
  const float* pos1 = (const float*)d_in[0];
  const float* pos2 = (const float*)d_in[1];
  const float* neg  = (const float*)d_in[2];
  float* ws  = (float*)d_ws;
  float* out = (float*)d_out;

  (void)in_sizes; (void)n_in; (void)out_size; (void)ws_size;

  zero_ws<<<1, THREADS, 0, stream>>>(ws, NROWS * 4);
  gram_stage1<<<NTILES * KSPLITS, THREADS, 0, stream>>>(pos1, pos2, neg, ws);
  finalize<<<1, THREADS, 0, stream>>>(ws, out);
}